// pixelwise_XDEDLoss_60636348285184
// MI455X (gfx1250) — compile-verified
//
#include <hip/hip_runtime.h>
#include <hip/hip_bf16.h>

// ---- types for WMMA ----
typedef __attribute__((ext_vector_type(2))) float v2f;   // A/B of 16x16x4 f32 (64 vals / 32 lanes)
typedef __attribute__((ext_vector_type(8))) float v8f;   // C/D 16x16 f32

#define C_NUM    19
#define SEGW     20          // ws row width: dims 0..18 + count at col 19
#define IGNORE_L 255

// -------------------------------------------------------------------------
// zero scratch (20x20 sums/counts) + output scalar; runs every launch so the
// whole pipeline is deterministic under graph replay.
// -------------------------------------------------------------------------
__global__ void xded_zero(float* __restrict__ ws, float* __restrict__ out) {
    int i = threadIdx.x;
    if (i < SEGW * SEGW) ws[i] = 0.0f;
    if (i == 0) out[0] = 0.0f;
}

// -------------------------------------------------------------------------
// Phase 1: sums[c][k] and counts[c] via one-hot matmul on V_WMMA_F32_16X16X4_F32.
// Per group: 4 pixels (K=4), 4 WMMAs covering the 2x2 tile:
//   [classes 0-15 | 16-18] x [dims 0-15 | dims 16-18 + ones->counts]
// VGPR layout (ISA 7.12.2, 32-bit 16x4): K = 2*(lane>=16) + vgpr_index;
// A rows M = lane%16, B cols N = lane%16; D: M = r + 8*(lane>=16), N = lane%16.
// B_hi is built branch-free (clamped column + mask blend) to keep the hot
// loop free of exec-mask divergence.
// -------------------------------------------------------------------------
__global__ __launch_bounds__(256) void xded_phase1(
        const float* __restrict__ logits,  // [N,19]
        const int*   __restrict__ gts,     // [N]
        float*       __restrict__ sums,    // [20][20]
        int nGroups, int nWaves) {
    const int tid  = blockIdx.x * blockDim.x + threadIdx.x;
    const int wid  = tid >> 5;
    const int lane = threadIdx.x & 31;
    const int ln   = lane & 15;          // row (A) / col (B) within tile
    const int half = lane >> 4;          // selects K base 0 or 2

    // branch-free helpers for the hi B-tile
    const int   hcol  = (ln < 3) ? (16 + ln) : 0;   // clamped column index
    const float hmask = (ln < 3) ? 1.0f : 0.0f;     // keep dims 16..18
    const float hone  = (ln == 3) ? 1.0f : 0.0f;    // ones column -> counts

    v8f dll = {}, dlh = {}, dhl = {}, dhh = {};

    // two pixel-groups per trip: 8 loads issued ahead of 8 WMMAs
    for (int g = wid * 2; g < nGroups; g += nWaves * 2) {
        #pragma unroll
        for (int u = 0; u < 2; ++u) {
            const int       gg = g + u;
            const int       pk = gg * 4 + half * 2;        // pixel for K=half*2 (.x), +1 (.y)
            const long long b0 = (long long)pk * C_NUM;
            const long long b1 = b0 + C_NUM;

            // B_lo: dims 0..15
            v2f blo;
            blo.x = logits[b0 + ln];
            blo.y = logits[b1 + ln];

            // B_hi: cols 0..2 = dims 16..18, col 3 = 1.0 (counts), rest 0 (branch-free)
            v2f bhi;
            bhi.x = logits[b0 + hcol] * hmask + hone;
            bhi.y = logits[b1 + hcol] * hmask + hone;

            // A: one-hot of label (255-ignored pixels match no row -> contribute 0)
            const int s0 = gts[pk], s1 = gts[pk + 1];
            v2f alo, ahi;
            alo.x = (s0 == ln)      ? 1.0f : 0.0f;
            alo.y = (s1 == ln)      ? 1.0f : 0.0f;
            ahi.x = (s0 == 16 + ln) ? 1.0f : 0.0f;
            ahi.y = (s1 == 16 + ln) ? 1.0f : 0.0f;

            // EXEC all-ones: loop bound is wave-uniform, no divergent branches above.
            dll = __builtin_amdgcn_wmma_f32_16x16x4_f32(false, alo, false, blo, (short)0, dll, false, false);
            dlh = __builtin_amdgcn_wmma_f32_16x16x4_f32(false, alo, false, bhi, (short)0, dlh, false, false);
            dhl = __builtin_amdgcn_wmma_f32_16x16x4_f32(false, ahi, false, blo, (short)0, dhl, false, false);
            dhh = __builtin_amdgcn_wmma_f32_16x16x4_f32(false, ahi, false, bhi, (short)0, dhh, false, false);
        }
    }

    // D layout: element (vgpr r, lane) -> M = r + 8*half, N = ln
    const int N = ln;
    #pragma unroll
    for (int r = 0; r < 8; ++r) {
        const int M = r + 8 * half;
        atomicAdd(&sums[M * SEGW + N], dll[r]);                       // classes 0-15, dims 0-15
        if (N < 3)       atomicAdd(&sums[M * SEGW + 16 + N], dlh[r]); // dims 16-18
        else if (N == 3) atomicAdd(&sums[M * SEGW + 19],     dlh[r]); // counts
        if (M < 3) {                                                  // classes 16-18
            atomicAdd(&sums[(16 + M) * SEGW + N], dhl[r]);
            if (N < 3)       atomicAdd(&sums[(16 + M) * SEGW + 16 + N], dhh[r]);
            else if (N == 3) atomicAdd(&sums[(16 + M) * SEGW + 19],     dhh[r]);
        }
    }
}

// -------------------------------------------------------------------------
// Phase 2: per-pixel KL( softmax(mean[g]/T) || softmax(x/T) ), summed.
// Ignored pixels contribute exactly 0 (target == x) and are skipped.
// -------------------------------------------------------------------------
__global__ __launch_bounds__(256) void xded_phase2(
        const float* __restrict__ logits,
        const int*   __restrict__ gts,
        const float* __restrict__ sums,    // [20][20] sums + counts
        float*       __restrict__ out,
        int n, float scale) {
    __shared__ float means[SEGW * SEGW];
    __shared__ float red[256];

    // recompute means per block: 400 divides, negligible
    for (int i = threadIdx.x; i < SEGW * SEGW; i += blockDim.x) {
        const int c = i / SEGW, k = i % SEGW;
        float v = 0.0f;
        if (k < C_NUM) {
            const float cnt = sums[c * SEGW + 19];
            v = sums[i] / fmaxf(cnt, 1.0f);
        }
        means[i] = v;
    }
    __syncthreads();

    float acc = 0.0f;
    for (int p = blockIdx.x * blockDim.x + threadIdx.x; p < n;
         p += gridDim.x * blockDim.x) {
        const int g = gts[p];
        if (g == IGNORE_L || g >= C_NUM) continue;  // zero contribution

        const float* __restrict__ x = logits + (long long)p * C_NUM;
        const float* __restrict__ t = means + g * SEGW;

        float xv[C_NUM], tv[C_NUM];
        float mx = -1e30f, mt = -1e30f;
        #pragma unroll
        for (int k = 0; k < C_NUM; ++k) {
            xv[k] = x[k] * 0.5f;            // x / TEMP
            tv[k] = t[k] * 0.5f;
            mx = fmaxf(mx, xv[k]);
            mt = fmaxf(mt, tv[k]);
        }
        float sx = 0.0f, st = 0.0f;
        #pragma unroll
        for (int k = 0; k < C_NUM; ++k) {
            sx += __expf(xv[k] - mx);
            st += __expf(tv[k] - mt);
        }
        const float lx = mx + __logf(sx);   // logsumexp(x/T)
        const float lt = mt + __logf(st);   // logsumexp(t/T)
        float kl = 0.0f;
        #pragma unroll
        for (int k = 0; k < C_NUM; ++k) {
            const float lq = tv[k] - lt;    // log q_k
            const float lp = xv[k] - lx;    // log p_k
            kl += __expf(lq) * (lq - lp);
        }
        acc += kl;
    }

    // block tree reduction, one scaled atomic per block
    red[threadIdx.x] = acc;
    __syncthreads();
    for (int s = blockDim.x >> 1; s > 0; s >>= 1) {
        if (threadIdx.x < s) red[threadIdx.x] += red[threadIdx.x + s];
        __syncthreads();
    }
    if (threadIdx.x == 0) atomicAdd(out, red[0] * scale);
}

// -------------------------------------------------------------------------
extern "C" void kernel_launch(void* const* d_in, const int* in_sizes, int n_in,
                              void* d_out, int out_size, void* d_ws, size_t ws_size,
                              hipStream_t stream) {
    const float* logits = (const float*)d_in[0];   // [B,H,W,19] f32
    const int*   gts    = (const int*)d_in[1];     // [B,H,W] labels
    float*       out    = (float*)d_out;           // scalar f32
    float*       sums   = (float*)d_ws;            // 20x20 f32 sums/counts

    const int n       = in_sizes[1];               // pixel count N
    const int nGroups = n >> 2;                    // 4 pixels per wave-iter

    xded_zero<<<1, 512, 0, stream>>>(sums, out);

    const int p1_blocks = 256, p1_threads = 256;
    const int nWaves = p1_blocks * p1_threads / 32;
    xded_phase1<<<p1_blocks, p1_threads, 0, stream>>>(logits, gts, sums, nGroups, nWaves);

    const float scale = 4.0f / (float)n;           // TEMP^2 / N
    xded_phase2<<<2048, 256, 0, stream>>>(logits, gts, sums, out, n, scale);
}